// SimpleIIRFilter_49005576848082
// MI455X (gfx1250) — compile-verified
//
#include <hip/hip_runtime.h>

typedef float v2f __attribute__((ext_vector_type(2)));
typedef float v4f __attribute__((ext_vector_type(4)));
typedef float v8f __attribute__((ext_vector_type(8)));

#define NB    32
#define NF    256
#define NT    8192
#define BT    16                 // timesteps per WMMA block
#define NBLK  (NT / BT)          // 512 sequential blocks
#define UNITS (NF * (NB / 16))   // 512 independent wave-units

// One wave = one workgroup handles (filter f, 16-batch tile). Per 16-step block:
//   Y(16x16) = M(16x20) * Xwin(20x16)  +  W(16x4) * H(4x16)
// where M folds the 5-tap FIR into the IIR impulse-response matrix:
//   M[t][i] = sum_k b[k] * g[t-i+k]   (valid: k<=i, t-i+k>=0),  g = IIR impulse resp.
// Only the W*H WMMA is on the sequential critical path.
__global__ __launch_bounds__(32)
void iir_wmma_f32_kernel(const float* __restrict__ x,
                         const float* __restrict__ bc,
                         const float* __restrict__ ac,
                         float* __restrict__ y)
{
    __shared__ float gtab[16];

    const int lane = threadIdx.x & 31;
    const int unit = blockIdx.x;        // 512 single-wave workgroups

    const int  f   = unit >> 1;         // adjacent workgroups share filter coeffs
    const int  bt  = unit & 1;          // which 16-batch tile
    const int  row = lane & 15;
    const bool lo  = lane < 16;
    const int  kl  = lo ? 0 : 2;        // K-pair held by this lane half (A/B 32-bit layout)

    // Uniform per-wave coefficients
    const float b0 = bc[f*5+0], b1 = bc[f*5+1], b2 = bc[f*5+2],
                b3 = bc[f*5+3], b4 = bc[f*5+4];
    const float a0 = ac[f*4+0], a1 = ac[f*4+1], a2 = ac[f*4+2], a3 = ac[f*4+3];

    // ---- IIR impulse response g[m] = d(m) - (a3 g[m-1] + a2 g[m-2] + a1 g[m-3] + a0 g[m-4])
    {
        float p1 = 0.f, p2 = 0.f, p3 = 0.f, p4 = 0.f, gcap = 0.f;
        #pragma unroll
        for (int m = 0; m < 16; ++m) {
            float g = (m == 0 ? 1.0f : 0.0f) - (a3*p1 + a2*p2 + a1*p3 + a0*p4);
            if (m == row) gcap = g;
            p4 = p3; p3 = p2; p2 = p1; p1 = g;
        }
        gtab[row] = gcap;               // lanes m and m+16 write identical values
    }
    __syncthreads();                    // single-wave WG: barrier is a NOP + dscnt wait

    // ---- A-operands gathered from the g-table.
    // mA[r]  : steady-state M columns i = 4r+kl, 4r+kl+1
    // m0A[r] : block-0 variant (drop j = i-k < 4 taps so y[0..3] == 0 exactly)
    v2f mA[5], m0A[5];
    #pragma unroll
    for (int r = 0; r < 5; ++r) {
        #pragma unroll
        for (int s = 0; s < 2; ++s) {
            const int i = 4*r + kl + s;        // window column 0..19 (runtime via kl)
            float acc1 = 0.f, acc0 = 0.f;
            #pragma unroll
            for (int k = 0; k < 5; ++k) {
                const int   d  = row - i + k;  // g index (> 15 only when k > i, masked)
                const int   dc = d < 0 ? 0 : d;
                const float gv = gtab[dc];     // ds_load_b32, per-lane address
                const float bk = (k==0)?b0:(k==1)?b1:(k==2)?b2:(k==3)?b3:b4;
                const float tm = bk * gv;
                acc1 += ((d >= 0) && (k <= i    )) ? tm : 0.f;
                acc0 += ((d >= 0) && (k <= i - 4)) ? tm : 0.f;
            }
            mA[r][s]  = acc1;
            m0A[r][s] = acc0;
        }
    }

    // ---- State matrix via identity W[t][c] = -sum_{q<=c} a_q * g[t-(c-q)]
    v2f wA;
    #pragma unroll
    for (int s = 0; s < 2; ++s) {
        const int c = kl + s;
        float acc = 0.f;
        #pragma unroll
        for (int q = 0; q < 4; ++q) {
            const int   d  = row - c + q;
            const int   dc = d < 0 ? 0 : d;
            const float gv = gtab[dc];
            const float aq = (q==0)?a0:(q==1)?a1:(q==2)?a2:a3;
            acc += ((q <= c) && (d >= 0)) ? aq * gv : 0.f;
        }
        wA[s] = -acc;
    }

    // Row pointers: lane (and lane+16) own batch nb; (B,F,T) layout.
    const int nb = bt * 16 + row;
    const float* px = x + ((size_t)nb * NF + f) * (size_t)NT;
    float*       py = y + ((size_t)nb * NF + f) * (size_t)NT;

    // WGP-scope prefetch (pulls into ALL cache levels; builtin only reaches SE/SYS)
#define PREFETCH_WGP(PTR) asm volatile("global_prefetch_b8 %0, off" :: "v"(PTR) : "memory")

    v4f cr;                             // carry = x[t0-4 .. t0-1]
    v2f hb;                             // running state y[r-4..r-1]

    // ================= peeled block 0 (uses m0A; carry = 0, H = 0) =================
    {
        const v4f* pv = (const v4f*)px;
        v4f l0 = pv[0], l1 = pv[1], l2 = pv[2], l3 = pv[3];
        PREFETCH_WGP(px + 16 * BT);

        v2f wb1, wb2, wb3, wb4;         // wb0 would be the (zero) carry -> skipped
        wb1[0] = lo ? l0[0] : l0[2];  wb1[1] = lo ? l0[1] : l0[3];
        wb2[0] = lo ? l1[0] : l1[2];  wb2[1] = lo ? l1[1] : l1[3];
        wb3[0] = lo ? l2[0] : l2[2];  wb3[1] = lo ? l2[1] : l2[3];
        wb4[0] = lo ? l3[0] : l3[2];  wb4[1] = lo ? l3[1] : l3[3];

        v8f acc = {};
        acc = __builtin_amdgcn_wmma_f32_16x16x4_f32(false, m0A[1], false, wb1, (short)0, acc, false, false);
        acc = __builtin_amdgcn_wmma_f32_16x16x4_f32(false, m0A[2], false, wb2, (short)0, acc, false, false);
        acc = __builtin_amdgcn_wmma_f32_16x16x4_f32(false, m0A[3], false, wb3, (short)0, acc, false, false);
        acc = __builtin_amdgcn_wmma_f32_16x16x4_f32(false, m0A[4], false, wb4, (short)0, acc, false, false);

        const int off = lo ? 0 : 8;
        v4f s0, s1;
        s0[0]=acc[0]; s0[1]=acc[1]; s0[2]=acc[2]; s0[3]=acc[3];
        s1[0]=acc[4]; s1[1]=acc[5]; s1[2]=acc[6]; s1[3]=acc[7];
        __builtin_nontemporal_store(s0, (v4f*)(py + off));
        __builtin_nontemporal_store(s1, (v4f*)(py + off + 4));

        const int src = row + 16;
        float y12 = __shfl(acc[4], src, 32);
        float y13 = __shfl(acc[5], src, 32);
        float y14 = __shfl(acc[6], src, 32);
        float y15 = __shfl(acc[7], src, 32);
        hb[0] = lo ? y12 : y14;
        hb[1] = lo ? y13 : y15;
        cr = l3;
    }

    // Preload block 1 (register double buffer)
    v4f c0, c1, c2, c3;
    {
        const v4f* pv = (const v4f*)(px + BT);
        c0 = pv[0]; c1 = pv[1]; c2 = pv[2]; c3 = pv[3];
    }

    // ========================== steady-state blocks 1..511 =========================
    // unroll 2 so the double-buffer rotation renames away instead of emitting movs
    #pragma unroll 2
    for (int blk = 1; blk < NBLK; ++blk) {
        const int t0 = blk * BT;                 // output rows r = t0 .. t0+15

        // Issue NEXT block's loads now; consumed next iteration (full body of slack).
        v4f n0, n1, n2, n3;
        {
            const int tn = (blk + 1 < NBLK ? blk + 1 : blk) * BT;
            const v4f* pv = (const v4f*)(px + tn);
            n0 = pv[0]; n1 = pv[1]; n2 = pv[2]; n3 = pv[3];
        }

        {   // WGP-scope prefetch ~16 blocks ahead, clamped in-bounds
            int tp = t0 + 16 * BT; if (tp > NT - BT) tp = NT - BT;
            PREFETCH_WGP(px + tp);
        }

        // Pack raw x window into B-operand layout (10 cndmasks, no FIR math!)
        v2f wb0, wb1, wb2, wb3, wb4;
        wb0[0] = lo ? cr[0] : cr[2];  wb0[1] = lo ? cr[1] : cr[3];
        wb1[0] = lo ? c0[0] : c0[2];  wb1[1] = lo ? c0[1] : c0[3];
        wb2[0] = lo ? c1[0] : c1[2];  wb2[1] = lo ? c1[1] : c1[3];
        wb3[0] = lo ? c2[0] : c2[2];  wb3[1] = lo ? c2[1] : c2[3];
        wb4[0] = lo ? c3[0] : c3[2];  wb4[1] = lo ? c3[1] : c3[3];

        // Y = W*H + M*Xwin via 6x V_WMMA_F32_16X16X4_F32 (W*H first: critical path)
        v8f acc = {};
        acc = __builtin_amdgcn_wmma_f32_16x16x4_f32(false, wA,    false, hb,  (short)0, acc, false, false);
        acc = __builtin_amdgcn_wmma_f32_16x16x4_f32(false, mA[0], false, wb0, (short)0, acc, false, false);
        acc = __builtin_amdgcn_wmma_f32_16x16x4_f32(false, mA[1], false, wb1, (short)0, acc, false, false);
        acc = __builtin_amdgcn_wmma_f32_16x16x4_f32(false, mA[2], false, wb2, (short)0, acc, false, false);
        acc = __builtin_amdgcn_wmma_f32_16x16x4_f32(false, mA[3], false, wb3, (short)0, acc, false, false);
        acc = __builtin_amdgcn_wmma_f32_16x16x4_f32(false, mA[4], false, wb4, (short)0, acc, false, false);

        // Store D (streamed once, never re-read -> nontemporal b128 stores).
        const int off = lo ? 0 : 8;
        v4f s0, s1;
        s0[0]=acc[0]; s0[1]=acc[1]; s0[2]=acc[2]; s0[3]=acc[3];
        s1[0]=acc[4]; s1[1]=acc[5]; s1[2]=acc[6]; s1[3]=acc[7];
        __builtin_nontemporal_store(s0, (v4f*)(py + t0 + off));
        __builtin_nontemporal_store(s1, (v4f*)(py + t0 + off + 4));

        // Next-block state H = Y rows 12..15 (live in acc[4..7] of hi lanes)
        const int src = row + 16;
        float y12 = __shfl(acc[4], src, 32);
        float y13 = __shfl(acc[5], src, 32);
        float y14 = __shfl(acc[6], src, 32);
        float y15 = __shfl(acc[7], src, 32);
        hb[0] = lo ? y12 : y14;
        hb[1] = lo ? y13 : y15;

        cr = c3;                         // x[t0+12 .. t0+15] becomes next look-back
        c0 = n0; c1 = n1; c2 = n2; c3 = n3;   // rotate double buffer
    }
#undef PREFETCH_WGP
}

extern "C" void kernel_launch(void* const* d_in, const int* in_sizes, int n_in,
                              void* d_out, int out_size, void* d_ws, size_t ws_size,
                              hipStream_t stream) {
    const float* x = (const float*)d_in[0];
    const float* b = (const float*)d_in[1];
    const float* a = (const float*)d_in[2];
    float* y = (float*)d_out;

    // One wave per workgroup -> 512 workgroups spread across all WGPs
    dim3 grid(UNITS), block(32);
    iir_wmma_f32_kernel<<<grid, block, 0, stream>>>(x, b, a, y);
}